// LanguageModelLSTM1_1477468750437
// MI455X (gfx1250) — compile-verified
//
#include <hip/hip_runtime.h>
#include <hip/hip_bf16.h>
#include <math.h>

// ---------------- shapes ----------------
#define Bsz   64
#define Tlen  20
#define Vsz   20000
#define Esz   512
#define Hsz   512
#define Lsz   49

typedef __attribute__((ext_vector_type(2))) float v2f;
typedef __attribute__((ext_vector_type(8))) float v8f;

#define WMMA_F32(a, b, c) \
    __builtin_amdgcn_wmma_f32_16x16x4_f32(false, (a), false, (b), (short)0, (c), false, false)

// =====================================================================
// fp32 WMMA GEMM:  C[M,N] = A[M,K] @ W[N,K]^T (+bias[N]) (+length mask)
// One wave computes a 32x32 tile (2x2 grid of 16x16 WMMA sub-tiles) --
// all M/N in this workload are multiples of 32, all K multiples of 32.
// Per K-step of 4: 2 A-frags + 2 B-frags (1 KB) feed 4 WMMAs (8 KFLOP).
// A-frag (ISA 7.12.2, 32-bit A 16x4): lane<16 -> K=k,k+1 ; lane>=16 -> k+2,k+3
// D layout: vgpr r, lane<16 -> C[M=r][N=lane]; lane>=16 -> C[M=8+r][N=lane-16]
// a_row_div>0: A row m lives at (m/a_row_div)*a_bstride + (m%a_row_div)*K
// lengths!=0 : row m -> b=m/maskT, t=m%maskT; zero output where t>=lengths[b]
// =====================================================================
__global__ void gemm_wmma_f32(const float* __restrict__ A,
                              const float* __restrict__ W,
                              const float* __restrict__ bias,
                              float* __restrict__ C,
                              int M, int N, int K,
                              int a_row_div, long long a_bstride,
                              const int* __restrict__ lengths, int maskT) {
    int wave = blockIdx.x * (blockDim.x >> 5) + (threadIdx.x >> 5);
    int tiles_n = N >> 5;
    int tiles_m = M >> 5;
    if (wave >= tiles_m * tiles_n) return;          // wave-uniform exit
    int tm = wave / tiles_n;
    int tn = wave - tm * tiles_n;

    int lane = threadIdx.x & 31;
    int half = lane >> 4;        // 0: K=k,k+1   1: K=k+2,k+3
    int lr   = lane & 15;
    int koff = half << 1;

    long long am0 = (long long)tm * 32 + lr;
    long long am1 = am0 + 16;
    const float *Arow0, *Arow1;
    if (a_row_div > 0) {
        Arow0 = A + (am0 / a_row_div) * a_bstride + (am0 % a_row_div) * (long long)K;
        Arow1 = A + (am1 / a_row_div) * a_bstride + (am1 % a_row_div) * (long long)K;
    } else {
        Arow0 = A + am0 * (long long)K;
        Arow1 = A + am1 * (long long)K;
    }
    const float* Wrow0 = W + ((long long)tn * 32 + lr) * (long long)K;
    const float* Wrow1 = Wrow0 + (long long)16 * K;

    v8f acc00 = {}, acc01 = {}, acc10 = {}, acc11 = {};
    for (int k0 = 0; k0 < K; k0 += 32) {
        if (k0 + 32 < K) {   // prefetch next K-chunk of the streamed operand
            __builtin_prefetch(Wrow0 + k0 + 32, 0, 1);
            __builtin_prefetch(Wrow1 + k0 + 32, 0, 1);
        }
        #pragma unroll
        for (int kk = 0; kk < 32; kk += 4) {
            int k = k0 + kk + koff;
            v2f a0 = *(const v2f*)(Arow0 + k);
            v2f a1 = *(const v2f*)(Arow1 + k);
            v2f b0 = *(const v2f*)(Wrow0 + k);
            v2f b1 = *(const v2f*)(Wrow1 + k);
            acc00 = WMMA_F32(a0, b0, acc00);
            acc01 = WMMA_F32(a0, b1, acc01);
            acc10 = WMMA_F32(a1, b0, acc10);
            acc11 = WMMA_F32(a1, b1, acc11);
        }
    }

    // -------- epilogue: 4 sub-tiles --------
    int col0 = tn * 32 + lr;
    int col1 = col0 + 16;
    float bad0 = bias ? bias[col0] : 0.0f;
    float bad1 = bias ? bias[col1] : 0.0f;
    #pragma unroll
    for (int r = 0; r < 8; ++r) {
        int row0 = tm * 32 + r + (half << 3);
        int row1 = row0 + 16;
        float v00 = acc00[r] + bad0;
        float v01 = acc01[r] + bad1;
        float v10 = acc10[r] + bad0;
        float v11 = acc11[r] + bad1;
        if (lengths) {
            int b0_ = row0 / maskT, t0_ = row0 - b0_ * maskT;
            int b1_ = row1 / maskT, t1_ = row1 - b1_ * maskT;
            if (t0_ >= lengths[b0_]) { v00 = 0.0f; v01 = 0.0f; }
            if (t1_ >= lengths[b1_]) { v10 = 0.0f; v11 = 0.0f; }
        }
        C[(long long)row0 * N + col0] = v00;
        C[(long long)row0 * N + col1] = v01;
        C[(long long)row1 * N + col0] = v10;
        C[(long long)row1 * N + col1] = v11;
    }
}

// ---------------- f = features[:,:,0].reshape(B,512,49).transpose(0,2,1) ----
__global__ void build_fmat(const float* __restrict__ feat, float* __restrict__ fmat) {
    long long i = (long long)blockIdx.x * blockDim.x + threadIdx.x;
    if (i >= (long long)Bsz * Lsz * 512) return;
    int d = (int)(i & 511);
    long long m = i >> 9;
    int b = (int)(m / Lsz), l = (int)(m % Lsz);
    fmat[i] = feat[(long long)b * 512 * Lsz + (long long)d * Lsz + l];
}

// ---------------- row L2 normalize (rows of 512) ----------------
__global__ void rownorm512(float* __restrict__ X) {
    __shared__ float red[8];
    int m = blockIdx.x, tid = threadIdx.x;
    float* row = X + (long long)m * 512;
    float s = 0.f;
    for (int i = tid; i < 512; i += 256) { float v = row[i]; s += v * v; }
    for (int o = 16; o > 0; o >>= 1) s += __shfl_down(s, o, 32);
    if ((tid & 31) == 0) red[tid >> 5] = s;
    __syncthreads();
    if (tid == 0) {
        float t = 0.f;
        for (int w = 0; w < 8; ++w) t += red[w];
        red[0] = 1.0f / fmaxf(sqrtf(t), 1e-12f);
    }
    __syncthreads();
    float inv = red[0];
    for (int i = tid; i < 512; i += 256) row[i] *= inv;
}

// ---------------- we = embed_W[captions] ----------------
__global__ void gather_embed(const float* __restrict__ E, const int* __restrict__ cap,
                             float* __restrict__ we) {
    int row = blockIdx.x;                 // 0..B*T-1
    int tok = cap[row];
    const float* src = E + (long long)tok * Esz;
    float* dst = we + (long long)row * Esz;
    for (int i = threadIdx.x; i < Esz; i += blockDim.x) dst[i] = src[i];
}

// ---------------- attention: one block per (b,t) of B*(T-1) ----------------
__global__ void attention(const float* __restrict__ att_fea,
                          const float* __restrict__ att_h,
                          const float* __restrict__ att_bias,
                          const float* __restrict__ att_w,
                          const float* __restrict__ fv,
                          float* __restrict__ ctx) {
    __shared__ float hvec[512];
    __shared__ float score[Lsz];
    __shared__ float red[8];
    int bt = blockIdx.x;                  // b*(T-1)+t
    int b  = bt / (Tlen - 1);
    int tid = threadIdx.x;

    const float* hrow = att_h + (long long)bt * 512;
    for (int i = tid; i < 512; i += 256) hvec[i] = hrow[i];
    __syncthreads();

    for (int l = 0; l < Lsz; ++l) {
        const float* fr = att_fea + ((long long)b * Lsz + l) * 512;
        float bl = att_bias[l];
        float s = 0.f;
        for (int i = tid; i < 512; i += 256) {
            float v = fr[i] + hvec[i] + bl;
            v = v > 0.f ? v : 0.f;
            s += v * att_w[i];
        }
        for (int o = 16; o > 0; o >>= 1) s += __shfl_down(s, o, 32);
        if ((tid & 31) == 0) red[tid >> 5] = s;
        __syncthreads();
        if (tid == 0) {
            float t = 0.f;
            for (int w = 0; w < 8; ++w) t += red[w];
            score[l] = t;
        }
        __syncthreads();
    }
    if (tid == 0) {
        float mx = score[0];
        for (int l = 1; l < Lsz; ++l) mx = fmaxf(mx, score[l]);
        float sum = 0.f;
        for (int l = 0; l < Lsz; ++l) { float e = expf(score[l] - mx); score[l] = e; sum += e; }
        float inv = 1.0f / sum;
        for (int l = 0; l < Lsz; ++l) score[l] *= inv;
    }
    __syncthreads();
    float* crow = ctx + (long long)bt * 512;
    for (int i = tid; i < 512; i += 256) {
        float acc = 0.f;
        for (int l = 0; l < Lsz; ++l)
            acc += score[l] * fv[((long long)b * Lsz + l) * 512 + i];
        crow[i] = acc;
    }
}

// ---------------- inputs = concat([feas_seq, we], -1) ----------------
__global__ void build_inputs(const float* __restrict__ feas0,
                             const float* __restrict__ ctx,
                             const float* __restrict__ we,
                             float* __restrict__ inp) {
    long long i = (long long)blockIdx.x * blockDim.x + threadIdx.x;
    if (i >= (long long)Bsz * Tlen * 1024) return;
    int col = (int)(i & 1023);
    long long m = i >> 10;
    int b = (int)(m / Tlen), t = (int)(m % Tlen);
    float v;
    if (col < 512)
        v = (t == 0) ? feas0[(long long)b * 512 + col]
                     : ctx[((long long)b * (Tlen - 1) + (t - 1)) * 512 + col];
    else
        v = we[((long long)b * Tlen + t) * Esz + (col - 512)];
    inp[i] = v;
}

__global__ void zero_f(float* p, long long n) {
    long long i = (long long)blockIdx.x * blockDim.x + threadIdx.x;
    if (i < n) p[i] = 0.0f;
}

// ---------------- LSTM step elementwise (one launch per t) ----------------
__global__ void lstm_step(const float* __restrict__ xg,
                          const float* __restrict__ g,      // h @ W_hh^T (64x2048)
                          const float* __restrict__ b_hh,
                          const int* __restrict__ lengths, int t,
                          float* __restrict__ h, float* __restrict__ c,
                          float* __restrict__ Hall) {
    long long i = (long long)blockIdx.x * blockDim.x + threadIdx.x;
    if (i >= (long long)Bsz * Hsz) return;
    int j = (int)(i & 511);
    int b = (int)(i >> 9);
    const float* xr = xg + ((long long)b * Tlen + t) * (4 * Hsz);
    const float* gr = g + (long long)b * (4 * Hsz);
    float gi = xr[j]            + gr[j]            + b_hh[j];
    float gf = xr[Hsz + j]      + gr[Hsz + j]      + b_hh[Hsz + j];
    float gg = xr[2 * Hsz + j]  + gr[2 * Hsz + j]  + b_hh[2 * Hsz + j];
    float go = xr[3 * Hsz + j]  + gr[3 * Hsz + j]  + b_hh[3 * Hsz + j];
    float si = 1.0f / (1.0f + expf(-gi));
    float sf = 1.0f / (1.0f + expf(-gf));
    float so = 1.0f / (1.0f + expf(-go));
    float tg = tanhf(gg);
    float cv = c[i], hv = h[i];
    float cn = sf * cv + si * tg;
    float hn = so * tanhf(cn);
    bool m = t < lengths[b];
    hv = m ? hn : hv;
    cv = m ? cn : cv;
    h[i] = hv; c[i] = cv;
    Hall[((long long)b * Tlen + t) * Hsz + j] = hv;
}

// =====================================================================
static inline void launch_gemm(const float* A, const float* W, const float* bias,
                               float* C, int M, int N, int K,
                               int arow_div, long long abstride,
                               const int* lengths, int maskT, hipStream_t s) {
    int tiles = (M / 32) * (N / 32);
    int blocks = (tiles + 7) / 8;                 // 8 waves / 256-thread block
    gemm_wmma_f32<<<blocks, 256, 0, s>>>(A, W, bias, C, M, N, K,
                                         arow_div, abstride, lengths, maskT);
}

extern "C" void kernel_launch(void* const* d_in, const int* in_sizes, int n_in,
                              void* d_out, int out_size, void* d_ws, size_t ws_size,
                              hipStream_t stream) {
    const float* feature_0   = (const float*)d_in[0];
    const float* features    = (const float*)d_in[1];
    // d_in[2] eng_embedding: unused by reference
    const int*   captions    = (const int*)d_in[3];
    const int*   lengths     = (const int*)d_in[4];
    const float* embed_W     = (const float*)d_in[5];
    const float* W_ih        = (const float*)d_in[6];
    const float* W_hh        = (const float*)d_in[7];
    const float* b_ih        = (const float*)d_in[8];
    const float* b_hh        = (const float*)d_in[9];
    const float* fc_out_W    = (const float*)d_in[10];
    const float* fc_out_b    = (const float*)d_in[11];
    const float* att_vw_W    = (const float*)d_in[12];
    const float* att_hw_W    = (const float*)d_in[13];
    const float* att_bias    = (const float*)d_in[14];
    const float* att_w_W     = (const float*)d_in[15];
    const float* att_fc_v_W  = (const float*)d_in[16];
    const float* att_fc_v_b  = (const float*)d_in[17];
    const float* img_embed_W = (const float*)d_in[18];
    const float* img_embed_b = (const float*)d_in[19];
    float* out = (float*)d_out;

    // ---- workspace carve-up (floats) ----
    float* ws = (float*)d_ws;
    size_t off = 0;
    float* fmat    = ws + off; off += (size_t)Bsz * Lsz * 512;       // 3136x512
    float* fv      = ws + off; off += (size_t)Bsz * Lsz * 512;
    float* att_fea = ws + off; off += (size_t)Bsz * Lsz * 512;
    float* we      = ws + off; off += (size_t)Bsz * Tlen * Esz;      // 1280x512
    float* att_h   = ws + off; off += (size_t)Bsz * (Tlen - 1) * 512;
    float* ctx     = ws + off; off += (size_t)Bsz * (Tlen - 1) * 512;
    float* feas0   = ws + off; off += (size_t)Bsz * 512;
    float* inputs  = ws + off; off += (size_t)Bsz * Tlen * 1024;     // 1280x1024
    float* xg      = ws + off; off += (size_t)Bsz * Tlen * 4 * Hsz;  // 1280x2048
    float* gbuf    = ws + off; off += (size_t)Bsz * 4 * Hsz;
    float* hbuf    = ws + off; off += (size_t)Bsz * Hsz;
    float* cbuf    = ws + off; off += (size_t)Bsz * Hsz;
    float* Hall    = ws + off; off += (size_t)Bsz * Tlen * Hsz;
    (void)ws_size; (void)in_sizes; (void)n_in; (void)out_size;

    // 1. fmat[b*49+l, d] = features[b, d, 0, l]
    {
        long long n = (long long)Bsz * Lsz * 512;
        build_fmat<<<(unsigned)((n + 255) / 256), 256, 0, stream>>>(features, fmat);
    }
    // 2. fv = fmat @ att_fc_v_W^T + b   (3136 x 512, K=512)
    launch_gemm(fmat, att_fc_v_W, att_fc_v_b, fv, Bsz * Lsz, 512, 512,
                0, 0, nullptr, 0, stream);
    // 3. row normalize fv
    rownorm512<<<Bsz * Lsz, 256, 0, stream>>>(fv);
    // 4. att_fea = fv @ att_vw_W^T
    launch_gemm(fv, att_vw_W, nullptr, att_fea, Bsz * Lsz, 512, 512,
                0, 0, nullptr, 0, stream);
    // 5. we = embed_W[captions]
    gather_embed<<<Bsz * Tlen, 256, 0, stream>>>(embed_W, captions, we);
    // 6. att_h = we[:, :T-1] @ att_hw_W^T   (row remap: 19 rows per batch of 20)
    launch_gemm(we, att_hw_W, nullptr, att_h, Bsz * (Tlen - 1), 512, 512,
                Tlen - 1, (long long)Tlen * Esz, nullptr, 0, stream);
    // 7. attention -> ctx  (B*(T-1) blocks)
    attention<<<Bsz * (Tlen - 1), 256, 0, stream>>>(att_fea, att_h, att_bias,
                                                    att_w_W, fv, ctx);
    // 8. feas0 = feature_0 @ img_embed_W^T + b   (64 x 512, K=2048)
    launch_gemm(feature_0, img_embed_W, img_embed_b, feas0, Bsz, 512, 2048,
                0, 0, nullptr, 0, stream);
    // 9. inputs = concat([feas_seq, we], -1)
    {
        long long n = (long long)Bsz * Tlen * 1024;
        build_inputs<<<(unsigned)((n + 255) / 256), 256, 0, stream>>>(feas0, ctx, we, inputs);
    }
    // 10. xg = inputs @ W_ih^T + b_ih  (1280 x 2048, K=1024)
    launch_gemm(inputs, W_ih, b_ih, xg, Bsz * Tlen, 4 * Hsz, 1024,
                0, 0, nullptr, 0, stream);
    // 11. h = c = 0
    {
        long long n = (long long)Bsz * Hsz;
        zero_f<<<(unsigned)((n + 255) / 256), 256, 0, stream>>>(hbuf, n);
        zero_f<<<(unsigned)((n + 255) / 256), 256, 0, stream>>>(cbuf, n);
    }
    // 12. LSTM scan: 20 sequential (GEMM + elementwise) launches
    for (int t = 0; t < Tlen; ++t) {
        launch_gemm(hbuf, W_hh, nullptr, gbuf, Bsz, 4 * Hsz, Hsz,
                    0, 0, nullptr, 0, stream);
        long long n = (long long)Bsz * Hsz;
        lstm_step<<<(unsigned)((n + 255) / 256), 256, 0, stream>>>(
            xg, gbuf, b_hh, lengths, t, hbuf, cbuf, Hall);
    }
    // 13. out = mask * (Hall @ fc_out_W^T + fc_out_b)   (1280 x 20000, K=512)
    launch_gemm(Hall, fc_out_W, fc_out_b, out, Bsz * Tlen, Vsz, 512,
                0, 0, lengths, Tlen, stream);
}